// CopyBackProjection_14130442404023
// MI455X (gfx1250) — compile-verified
//
#include <hip/hip_runtime.h>

typedef __attribute__((ext_vector_type(2))) float v2f;
typedef __attribute__((ext_vector_type(8))) float v8f;

#define KOFF 27

enum { EPI_NONE = 0, EPI_SUB = 1, EPI_ADD = 2 };

// ---------------------------------------------------------------------------
// up[n, c] = x[n, c & 7]   (channel tiling, C_IN=8 -> C_OUT=32)
// ---------------------------------------------------------------------------
__global__ void upsample_kernel(const float* __restrict__ x,
                                float* __restrict__ up, int n) {
  int i = blockIdx.x * blockDim.x + threadIdx.x;
  if (i < n * 32) {
    int row = i >> 5;
    int c   = i & 31;
    up[i] = x[row * 8 + (c & 7)];
  }
}

// ---------------------------------------------------------------------------
// Sparse conv: out[v] = sum_k mask[k,v] * feat[nbr[k,v]] @ W[k]  (+/- aux)
// One wave = 16 voxels x COUT outputs via V_WMMA_F32_16X16X4_F32.
//   A (16x4 f32): lane L holds row M=L%16, K-columns (kb+2*(L/16), +1).
//   B (4x16 f32): lane L holds col N=L%16, K-rows  (kb+2*(L/16), +1).
//   D (16x16 f32, v8f): acc[j] = out[M = 8*(L/16)+j][N = L%16].
// W[k] (CIN*COUT <= 1024 floats) staged in LDS per block per k.
// Wave-uniform ballot skip when all 16 masks of this k-slice are zero.
// ---------------------------------------------------------------------------
template <int CIN, int COUT, int EPI>
__global__ __launch_bounds__(256) void sparse_conv_wmma(
    const float* __restrict__ feat,         // [n, CIN]
    const float* __restrict__ W,            // [KOFF, CIN, COUT]
    const int* __restrict__ nbr,            // [KOFF, n]
    const unsigned char* __restrict__ msk,  // [KOFF, n] (bool)
    const float* __restrict__ aux,          // [n, COUT] or null
    float* __restrict__ out,                // [n, COUT]
    int n) {
  constexpr int WAVES = 8;
  __shared__ float lds_w[CIN * COUT];

  const int lane = threadIdx.x & 31;
  const int wave = threadIdx.x >> 5;
  const int r    = lane & 15;  // row in tile (A) / output column (B,D)
  const int half = lane >> 4;  // selects K pair (A/B) and M half (D)
  const int tile = blockIdx.x * WAVES + wave;
  const bool tile_ok = (tile * 16) < n;              // wave-uniform
  const int voxel = min(tile * 16 + r, n - 1);       // clamped gather row

  v8f acc0 = {};
  v8f acc1 = {};

  for (int k = 0; k < KOFF; ++k) {
    // ---- stage W[k] into LDS (block-cooperative) ----
    for (int i = threadIdx.x; i < CIN * COUT; i += blockDim.x)
      lds_w[i] = W[(size_t)k * CIN * COUT + i];
    __syncthreads();

    if (tile_ok) {
      if (k + 1 < KOFF) {  // pull next k-slice's index/mask lines toward L2
        __builtin_prefetch(nbr + (size_t)(k + 1) * n + voxel, 0, 1);
        __builtin_prefetch(msk + (size_t)(k + 1) * n + voxel, 0, 1);
      }
      const int   nidx = nbr[(size_t)k * n + voxel];
      const float mval = msk[(size_t)k * n + voxel] ? 1.0f : 0.0f;
      // ballot is wave-uniform -> branch keeps EXEC all-ones for WMMA
      if (__ballot(mval != 0.0f) != 0ull) {
        const float* src = feat + (size_t)nidx * CIN;
#pragma unroll
        for (int kb = 0; kb < CIN; kb += 4) {
          const int kr = kb + 2 * half;  // first K-row held by this lane
          v2f a = *(const v2f*)(src + kr);  // 8B-aligned gather
          a.x *= mval;
          a.y *= mval;
          v2f b0;
          b0.x = (COUT >= 16 || r < COUT) ? lds_w[kr * COUT + r] : 0.0f;
          b0.y = (COUT >= 16 || r < COUT) ? lds_w[(kr + 1) * COUT + r] : 0.0f;
          acc0 = __builtin_amdgcn_wmma_f32_16x16x4_f32(
              false, a, false, b0, (short)0, acc0, false, false);
          if (COUT > 16) {
            v2f b1;
            b1.x = lds_w[kr * COUT + 16 + r];
            b1.y = lds_w[(kr + 1) * COUT + 16 + r];
            acc1 = __builtin_amdgcn_wmma_f32_16x16x4_f32(
                false, a, false, b1, (short)0, acc1, false, false);
          }
        }
      }
    }
    __syncthreads();  // protect lds_w before next k overwrites it
  }

  // ---- epilogue: D layout -> global, with optional fused aux -/+ ----
  if (tile_ok) {
#pragma unroll
    for (int j = 0; j < 8; ++j) {
      const int row = tile * 16 + half * 8 + j;
      if (row < n) {
        if (COUT > 16 || r < COUT) {
          float v = acc0[j];
          if (EPI == EPI_SUB) v = aux[(size_t)row * COUT + r] - v;
          if (EPI == EPI_ADD) v = aux[(size_t)row * COUT + r] + v;
          out[(size_t)row * COUT + r] = v;
        }
        if (COUT > 16) {
          const int c = 16 + r;
          float v = acc1[j];
          if (EPI == EPI_SUB) v = aux[(size_t)row * COUT + c] - v;
          if (EPI == EPI_ADD) v = aux[(size_t)row * COUT + c] + v;
          out[(size_t)row * COUT + c] = v;
        }
      }
    }
  }
}

// ---------------------------------------------------------------------------
extern "C" void kernel_launch(void* const* d_in, const int* in_sizes, int n_in,
                              void* d_out, int out_size, void* d_ws,
                              size_t ws_size, hipStream_t stream) {
  const float*         x   = (const float*)d_in[0];
  const int*           nbr = (const int*)d_in[1];
  const unsigned char* msk = (const unsigned char*)d_in[2];
  const float*         Wd0 = (const float*)d_in[3];
  const float*         Wd1 = (const float*)d_in[4];
  const float*         Wd2 = (const float*)d_in[5];
  const float*         Wu0 = (const float*)d_in[6];
  const float*         Wu1 = (const float*)d_in[7];
  const float*         Wu2 = (const float*)d_in[8];

  const int n = in_sizes[0] / 8;  // x is [n, 8]

  float* up   = (float*)d_ws;                 // [n, 32] kept for final add
  float* bufB = up + (size_t)n * 32;          // [n, 32] ping buffer
  float* b8a  = bufB + (size_t)n * 32;        // [n, 8]
  float* b8b  = b8a + (size_t)n * 8;          // [n, 8]
  float* outp = (float*)d_out;                // [n, 32]

  {
    int total = n * 32;
    upsample_kernel<<<(total + 255) / 256, 256, 0, stream>>>(x, up, n);
  }

  const int tiles  = (n + 15) / 16;
  const int blocks = (tiles + 7) / 8;

  // d1 = conv(up, Wd0)            32 -> 32
  sparse_conv_wmma<32, 32, EPI_NONE>
      <<<blocks, 256, 0, stream>>>(up, Wd0, nbr, msk, nullptr, bufB, n);
  // d2 = conv(d1, Wd1)            32 -> 8
  sparse_conv_wmma<32, 8, EPI_NONE>
      <<<blocks, 256, 0, stream>>>(bufB, Wd1, nbr, msk, nullptr, b8a, n);
  // r = x - conv(d2, Wd2)          8 -> 8  (fused residual)
  sparse_conv_wmma<8, 8, EPI_SUB>
      <<<blocks, 256, 0, stream>>>(b8a, Wd2, nbr, msk, x, b8b, n);
  // r1 = conv(r, Wu0)              8 -> 8
  sparse_conv_wmma<8, 8, EPI_NONE>
      <<<blocks, 256, 0, stream>>>(b8b, Wu0, nbr, msk, nullptr, b8a, n);
  // r2 = conv(r1, Wu1)             8 -> 32
  sparse_conv_wmma<8, 32, EPI_NONE>
      <<<blocks, 256, 0, stream>>>(b8a, Wu1, nbr, msk, nullptr, bufB, n);
  // out = up + conv(r2, Wu2)      32 -> 32 (fused add)
  sparse_conv_wmma<32, 32, EPI_ADD>
      <<<blocks, 256, 0, stream>>>(bufB, Wu2, nbr, msk, up, outp, n);
}